// NonLocalBlock_71485435674645
// MI455X (gfx1250) — compile-verified
//
#include <hip/hip_runtime.h>
#include <hip/hip_bf16.h>
#include <math.h>
#include <stdint.h>

// ---------------- problem constants (from reference) ----------------
#define BATCH 4
#define CCH   64          // channels
#define HH    256
#define WW    256
#define PBLK  4           // block size p
#define PH    64          // blocks per side = H/p
#define NBLK  4096        // PH*PH
#define PP    16          // p*p
#define KNN   4           // topk
#define INV_T 1000.0f     // 1/TEMP

typedef __attribute__((ext_vector_type(2))) float v2f;
typedef __attribute__((ext_vector_type(8))) float v8f;

// low 32 bits of a generic pointer to LDS == wave-relative LDS byte offset
__device__ __forceinline__ uint32_t lds_off(const void* p) {
    return (uint32_t)(uintptr_t)p;
}

// CDNA5 async global->LDS copy, 16 bytes per lane (ASYNCcnt tracked)
__device__ __forceinline__ void async_copy_b128(uint32_t lds_dst, const void* gsrc) {
    asm volatile("global_load_async_to_lds_b128 %0, %1, off"
                 :: "v"(lds_dst), "v"(gsrc)
                 : "memory");
}
__device__ __forceinline__ void wait_async0() {
    asm volatile("s_wait_asynccnt 0x0" ::: "memory");
}

// =====================================================================
// Kernel 1: 4x4 average pooling -> pooled descriptors [b][block][c]
// =====================================================================
__global__ __launch_bounds__(256)
void nlb_pool_kernel(const float* __restrict__ target,
                     const float* __restrict__ ref,
                     float* __restrict__ pt, float* __restrict__ pr)
{
    int t = blockIdx.x * blockDim.x + threadIdx.x;
    const int per = BATCH * NBLK * CCH;
    if (t >= 2 * per) return;
    int sel = t / per;
    int r   = t - sel * per;
    int b   = r / (NBLK * CCH);
    int blk = (r / CCH) % NBLK;
    int c   = r % CCH;
    int bh = blk >> 6, bw = blk & 63;
    const float* src = (sel ? ref : target) + (((size_t)b * CCH + c) * HH) * WW;
    float s = 0.0f;
    #pragma unroll
    for (int y = 0; y < PBLK; ++y) {
        const float* row = src + (size_t)(bh * PBLK + y) * WW + bw * PBLK;
        #pragma unroll
        for (int x = 0; x < PBLK; ++x) s += row[x];
    }
    float* dst = sel ? pr : pt;
    dst[r] = s * (1.0f / 16.0f);
}

// =====================================================================
// Kernel 2: squared norms of pooled descriptors
// =====================================================================
__global__ __launch_bounds__(256)
void nlb_norm_kernel(const float* __restrict__ pt, const float* __restrict__ pr,
                     float* __restrict__ tn, float* __restrict__ rn)
{
    int t = blockIdx.x * blockDim.x + threadIdx.x;
    const int per = BATCH * NBLK;
    if (t >= 2 * per) return;
    int sel = t / per;
    int i   = t - sel * per;
    const float* p = (sel ? pr : pt) + (size_t)i * CCH;
    float s = 0.0f;
    #pragma unroll
    for (int c = 0; c < CCH; ++c) s += p[c] * p[c];
    (sel ? rn : tn)[i] = s;
}

// ------- per-lane sorted top-4 insertion (stable: strict <) -------
__device__ __forceinline__ void ins4(float v, int i, float bv[4], int bi[4])
{
    if (v < bv[3]) {
        bv[3] = v; bi[3] = i;
        #pragma unroll
        for (int t = 3; t > 0; --t) {
            if (bv[t] < bv[t - 1]) {
                float tv = bv[t]; bv[t] = bv[t - 1]; bv[t - 1] = tv;
                int   ti = bi[t]; bi[t] = bi[t - 1]; bi[t - 1] = ti;
            }
        }
    }
}

// =====================================================================
// Kernel 3: fused distance GEMM (f32 WMMA 16x16x4) + top-4 per column.
//  One wave handles 16 ref-block columns; loops over 256 row tiles.
//  All 16 A-fragments prefetched per tile; dual accumulators break the
//  WMMA D->C dependency chain so loads/WMMAs pipeline.
// =====================================================================
__global__ __launch_bounds__(32)
void nlb_dist_topk_kernel(const float* __restrict__ pt, const float* __restrict__ pr,
                          const float* __restrict__ tn, const float* __restrict__ rn,
                          int* __restrict__ topi)
{
    int bid = blockIdx.x;
    int b  = bid >> 8;          // 256 j-tiles per batch
    int jt = bid & 255;
    int tid = threadIdx.x;
    int lr = tid & 15;          // fragment row/col selector
    int hi = tid >> 4;          // K half selector

    const float* ptb = pt + (size_t)b * NBLK * CCH;
    const float* prb = pr + (size_t)b * NBLK * CCH;

    // Preload B fragments for this 16-column tile: B[k][j], j = jt*16+lr.
    int j = jt * 16 + lr;
    v2f bf[16];
    #pragma unroll
    for (int kb = 0; kb < 16; ++kb) {
        const float* p = prb + (size_t)j * CCH + kb * 4 + hi * 2;
        bf[kb].x = p[0]; bf[kb].y = p[1];
    }
    float rnj = rn[b * NBLK + j];

    float bv[4] = {3.4e38f, 3.4e38f, 3.4e38f, 3.4e38f};
    int   bi4[4] = {0, 0, 0, 0};

    for (int it = 0; it < NBLK / 16; ++it) {
        // prefetch all 16 A fragments (16 independent global_load_b64)
        const float* rowp = ptb + (size_t)(it * 16 + lr) * CCH + hi * 2;
        v2f a[16];
        #pragma unroll
        for (int kb = 0; kb < 16; ++kb) {
            a[kb].x = rowp[kb * 4];
            a[kb].y = rowp[kb * 4 + 1];
        }
        v8f acc0 = {}, acc1 = {};
        #pragma unroll
        for (int kb = 0; kb < 16; kb += 2) {
            acc0 = __builtin_amdgcn_wmma_f32_16x16x4_f32(
                false, a[kb],     false, bf[kb],     (short)0, acc0, false, false);
            acc1 = __builtin_amdgcn_wmma_f32_16x16x4_f32(
                false, a[kb + 1], false, bf[kb + 1], (short)0, acc1, false, false);
        }
        // lane owns column j, rows i = it*16 + hi*8 + r
        int ibase = it * 16 + hi * 8;
        const float* tnb = tn + b * NBLK + ibase;
        #pragma unroll
        for (int r = 0; r < 8; ++r) {
            float d2 = tnb[r] + rnj - 2.0f * (acc0[r] + acc1[r]);
            ins4(d2, ibase + r, bv, bi4);
        }
    }

    // merge the two half-lanes owning the same column (lane L <-> L+16)
    float ov[4]; int oi[4];
    #pragma unroll
    for (int t = 0; t < 4; ++t) {
        ov[t] = __shfl_xor(bv[t], 16, 32);
        oi[t] = __shfl_xor(bi4[t], 16, 32);
    }
    float mv[4]; int mi[4];
    int pa = 0, pb = 0;
    #pragma unroll
    for (int t = 0; t < 4; ++t) {
        bool ta = (bv[pa] < ov[pb]) || (bv[pa] == ov[pb] && bi4[pa] < oi[pb]);
        if (ta) { mv[t] = bv[pa]; mi[t] = bi4[pa]; ++pa; }
        else    { mv[t] = ov[pb]; mi[t] = oi[pb]; ++pb; }
    }
    (void)mv;
    if (tid < 16) {
        int* o = topi + (size_t)(b * NBLK + jt * 16 + tid) * KNN;
        #pragma unroll
        for (int t = 0; t < KNN; ++t) o[t] = mi[t];
    }
}

// =====================================================================
// Kernel 4: per-block attention.  One wave per (b, n).
//   Tiles staged into LDS via GLOBAL_LOAD_ASYNC_TO_LDS_B128 (ASYNCcnt),
//   then: cross = Ft^T(16x64) . R(64x64) -> sharp softmax -> out = A.S^T
//   both GEMMs via f32 WMMA 16x16x4.
// =====================================================================
__global__ __launch_bounds__(32)
void nlb_attend_kernel(const float* __restrict__ target,
                       const float* __restrict__ ref,
                       const float* __restrict__ ref_align,
                       const int* __restrict__ topi,
                       float* __restrict__ out)
{
    __shared__ float sFt[CCH * PP];      // [c][s]  4 KB
    __shared__ float sR[CCH * KNN * PP]; // [c][m] 16 KB
    __shared__ float sA[CCH * KNN * PP]; // [c][m] 16 KB
    __shared__ float sS[PP * KNN * PP];  // [s][m]  4 KB
    __shared__ float sTn[PP];
    __shared__ float sRn[KNN * PP];

    int tid = threadIdx.x;
    int b = blockIdx.x >> 12;
    int n = blockIdx.x & (NBLK - 1);
    int bh = n >> 6, bw = n & 63;

    const size_t plane = (size_t)CCH * HH * WW;
    const float* tg = target    + (size_t)b * plane;
    const float* rf = ref       + (size_t)b * plane;
    const float* ra = ref_align + (size_t)b * plane;

    // ---- async-stage target block: quads of 4 pixels (one patch row) ----
    // quad q: c = q>>2, sq = q&3 ; global row (bh*4+sq), cols bw*4..bw*4+3
    #pragma unroll
    for (int q = tid; q < CCH * PP / 4; q += 32) {
        int c = q >> 2, sq = q & 3;
        const float* g = tg + ((size_t)c * HH + bh * PBLK + sq) * WW + bw * PBLK;
        async_copy_b128(lds_off(&sFt[c * PP + sq * 4]), g);
    }
    // ---- async-stage k neighbor blocks of ref / ref_align ----
    #pragma unroll
    for (int kk = 0; kk < KNN; ++kk) {
        int jn = topi[(size_t)(b * NBLK + n) * KNN + kk];
        int jh = jn >> 6, jw = jn & 63;
        #pragma unroll
        for (int q = tid; q < CCH * PP / 4; q += 32) {
            int c = q >> 2, sq = q & 3;
            size_t gofs = ((size_t)c * HH + jh * PBLK + sq) * WW + jw * PBLK;
            uint32_t l = (uint32_t)(c * (KNN * PP) + kk * PP + sq * 4);
            async_copy_b128(lds_off(&sR[l]), rf + gofs);
            async_copy_b128(lds_off(&sA[l]), ra + gofs);
        }
    }
    wait_async0();
    __syncthreads();

    // pixel norms
    if (tid < PP) {
        float s = 0.0f;
        #pragma unroll
        for (int c = 0; c < CCH; ++c) { float v = sFt[c * PP + tid]; s += v * v; }
        sTn[tid] = s;
    }
    for (int m = tid; m < KNN * PP; m += 32) {
        float s = 0.0f;
        #pragma unroll
        for (int c = 0; c < CCH; ++c) { float v = sR[c * (KNN * PP) + m]; s += v * v; }
        sRn[m] = s;
    }
    __syncthreads();

    int lr = tid & 15;
    int hi = tid >> 4;

    // GEMM1: cross[s][m] = sum_c Ft[c][s] * R[c][m]; logits into sS
    #pragma unroll
    for (int mt = 0; mt < 4; ++mt) {
        v8f acc = {};
        #pragma unroll
        for (int kb = 0; kb < 16; ++kb) {
            int k = kb * 4 + hi * 2;
            v2f a;  a.x = sFt[k * PP + lr];
                    a.y = sFt[(k + 1) * PP + lr];
            v2f bm; bm.x = sR[k * (KNN * PP) + mt * 16 + lr];
                    bm.y = sR[(k + 1) * (KNN * PP) + mt * 16 + lr];
            acc = __builtin_amdgcn_wmma_f32_16x16x4_f32(
                false, a, false, bm, (short)0, acc, false, false);
        }
        int m = mt * 16 + lr;
        float rm = sRn[m];
        #pragma unroll
        for (int r = 0; r < 8; ++r) {
            int s = hi * 8 + r;
            float d2 = sTn[s] + rm - 2.0f * acc[r];
            float d  = sqrtf(fmaxf(d2, 1e-30f));
            sS[s * (KNN * PP) + m] = -d * INV_T;
        }
    }
    __syncthreads();

    // softmax over m (64) per pixel row s
    if (tid < PP) {
        float* row = sS + tid * (KNN * PP);
        float mx = -3.4e38f;
        #pragma unroll
        for (int m = 0; m < KNN * PP; ++m) mx = fmaxf(mx, row[m]);
        float sum = 0.0f;
        #pragma unroll
        for (int m = 0; m < KNN * PP; ++m) { float e = __expf(row[m] - mx); row[m] = e; sum += e; }
        float inv = 1.0f / sum;
        #pragma unroll
        for (int m = 0; m < KNN * PP; ++m) row[m] *= inv;
    }
    __syncthreads();

    // GEMM2: out[c][s] = sum_m A[c][m] * S[s][m]; scatter into output image
    float* op = out + (size_t)b * plane;
    #pragma unroll
    for (int ct = 0; ct < 4; ++ct) {
        v8f acc = {};
        #pragma unroll
        for (int kb = 0; kb < 16; ++kb) {
            int k = kb * 4 + hi * 2;
            v2f a;  a.x = sA[(ct * 16 + lr) * (KNN * PP) + k];
                    a.y = sA[(ct * 16 + lr) * (KNN * PP) + k + 1];
            v2f bm; bm.x = sS[lr * (KNN * PP) + k];
                    bm.y = sS[lr * (KNN * PP) + k + 1];
            acc = __builtin_amdgcn_wmma_f32_16x16x4_f32(
                false, a, false, bm, (short)0, acc, false, false);
        }
        int s = lr;
        #pragma unroll
        for (int r = 0; r < 8; ++r) {
            int c = ct * 16 + hi * 8 + r;
            op[((size_t)c * HH + bh * PBLK + (s >> 2)) * WW + bw * PBLK + (s & 3)] = acc[r];
        }
    }
}

// =====================================================================
extern "C" void kernel_launch(void* const* d_in, const int* in_sizes, int n_in,
                              void* d_out, int out_size, void* d_ws, size_t ws_size,
                              hipStream_t stream)
{
    const float* target    = (const float*)d_in[0];
    const float* ref       = (const float*)d_in[1];
    const float* ref_align = (const float*)d_in[2];
    float* out = (float*)d_out;

    // workspace layout (~8.6 MB)
    float* pt = (float*)d_ws;                         // B*N*C
    float* pr = pt + (size_t)BATCH * NBLK * CCH;      // B*N*C
    float* tn = pr + (size_t)BATCH * NBLK * CCH;      // B*N
    float* rn = tn + (size_t)BATCH * NBLK;            // B*N
    int* topi = (int*)(rn + (size_t)BATCH * NBLK);    // B*N*K

    {
        int total = 2 * BATCH * NBLK * CCH;
        nlb_pool_kernel<<<(total + 255) / 256, 256, 0, stream>>>(target, ref, pt, pr);
    }
    {
        int total = 2 * BATCH * NBLK;
        nlb_norm_kernel<<<(total + 255) / 256, 256, 0, stream>>>(pt, pr, tn, rn);
    }
    nlb_dist_topk_kernel<<<BATCH * (NBLK / 16), 32, 0, stream>>>(pt, pr, tn, rn, topi);
    nlb_attend_kernel<<<BATCH * NBLK, 32, 0, stream>>>(target, ref, ref_align, topi, out);
}